// Attention_40424232190129
// MI455X (gfx1250) — compile-verified
//
#include <hip/hip_runtime.h>
#include <hip/hip_bf16.h>

typedef __attribute__((ext_vector_type(2))) float v2f;
typedef __attribute__((ext_vector_type(8))) float v8f;

#define BB 64
#define TT 1000
#define TPAD 1024
#define QD 1024
#define MD 512
#define AD 128
#define NF 32
#define KS 31
#define CPAD 15
#define LSTRIDE 66   // LDS row stride (floats): (2m+k) mod 64 -> conflict-free, 8B aligned

// ---------------- Stage 0: location conv  awc[B,1,T] -> locfeat[B,T,32] ----
__global__ __launch_bounds__(256) void conv_kernel(
    const float* __restrict__ awc, const float* __restrict__ conv_w,
    float* __restrict__ locfeat)
{
  __shared__ float xs[256 + KS - 1];
  __shared__ float wsd[NF * KS];
  const int b   = blockIdx.y;
  const int t0  = blockIdx.x * 256;
  const int tid = threadIdx.x;
  for (int i = tid; i < NF * KS; i += 256) wsd[i] = conv_w[i];
  for (int i = tid; i < 256 + KS - 1; i += 256) {
    int t = t0 - CPAD + i;
    xs[i] = (t >= 0 && t < TT) ? awc[b * TT + t] : 0.f;
  }
  __syncthreads();
  int t = t0 + tid;
  if (t < TT) {
    float* out = &locfeat[((size_t)b * TT + t) * NF];
    for (int f = 0; f < NF; ++f) {
      float s = 0.f;
      #pragma unroll
      for (int k = 0; k < KS; ++k) s += wsd[f * KS + k] * xs[tid + k];
      out[f] = s;
    }
  }
}

// ---------------- Stage 1: qcomb[b,a] = q@Wq^T + bq + bm + Wloc.conv_b ----
__global__ __launch_bounds__(128) void qproj_kernel(
    const float* __restrict__ query, const float* __restrict__ Wq,
    const float* __restrict__ bq, const float* __restrict__ bm,
    const float* __restrict__ conv_b, const float* __restrict__ Wloc,
    float* __restrict__ qcomb)
{
  __shared__ float qs[QD];
  const int b = blockIdx.x;
  const int a = threadIdx.x;
  for (int i = a; i < QD; i += 128) qs[i] = query[b * QD + i];
  __syncthreads();
  float s = bq[a] + bm[a];
  for (int f = 0; f < NF; ++f) s += Wloc[a * NF + f] * conv_b[f];
  const float* wrow = &Wq[(size_t)a * QD];
  float acc = 0.f;
  for (int k = 0; k < QD; ++k) acc += qs[k] * wrow[k];
  qcomb[b * AD + a] = s + acc;
}

// ---------------- Stage 2: WMMA energies -----------------------------------
// Block = 256 threads (8 waves). Block computes energies for one b and a
// 64-row tile of T.  energies[t] = Wv . tanh( mem@Wm^T + locfeat@Wloc^T + qcomb ) + bv
__global__ __launch_bounds__(256) void energies_kernel(
    const float* __restrict__ memory, const float* __restrict__ Wm,
    const float* __restrict__ locfeat, const float* __restrict__ Wloc,
    const float* __restrict__ qcomb, const float* __restrict__ Wv,
    const float* __restrict__ bvp, float* __restrict__ energies)
{
  __shared__ float As[64 * LSTRIDE];    // A tile: [m][k]
  __shared__ float Bs[AD * LSTRIDE];    // B tile transposed: [n][k]
  __shared__ float ener[64];

  const int b    = blockIdx.y;
  const int t0   = blockIdx.x * 64;
  const int tid  = threadIdx.x;
  const int lane = tid & 31;
  const int wave = tid >> 5;            // 0..7 -> n-range
  const int n0   = wave * 16;
  const int lm   = lane & 15;           // m (for A) / n (for B) within 16
  const int kh   = (lane >> 4) << 1;    // 0 or 2: K half handled by this lane

  if (tid < 64) ener[tid] = bvp[0];

  v8f acc[4] = {};                      // 4 M-tiles (rows t0+0..63), cols n0..n0+15

  // ---- main K loop: memory(64xK) @ Wm^T(KxN), K=512 in chunks of 64 ----
  for (int kc = 0; kc < MD; kc += 64) {
    __syncthreads();
    #pragma unroll
    for (int i = 0; i < 16; ++i) {      // As: 64x64
      int idx = tid + i * 256;
      int row = idx >> 6, col = idx & 63;
      int t = t0 + row;
      As[row * LSTRIDE + col] = (t < TT)
          ? memory[((size_t)b * TT + t) * MD + kc + col] : 0.f;
    }
    #pragma unroll
    for (int i = 0; i < 32; ++i) {      // Bs[n][k] = Wm[n][kc+k]: 128x64
      int idx = tid + i * 256;
      int n = idx >> 6, col = idx & 63;
      Bs[n * LSTRIDE + col] = Wm[(size_t)n * MD + kc + col];
    }
    __syncthreads();
    for (int kb = 0; kb < 64; kb += 4) {
      // B operand: VGPR0 = row K=kb+kh, VGPR1 = row K=kb+kh+1 (per-lane)
      v2f bvec = *reinterpret_cast<const v2f*>(&Bs[(n0 + lm) * LSTRIDE + kb + kh]);
      #pragma unroll
      for (int tile = 0; tile < 4; ++tile) {
        v2f avec = *reinterpret_cast<const v2f*>(&As[(tile * 16 + lm) * LSTRIDE + kb + kh]);
        acc[tile] = __builtin_amdgcn_wmma_f32_16x16x4_f32(
            false, avec, false, bvec, (short)0, acc[tile], false, false);
      }
    }
  }

  // ---- fused loc projection: locfeat(64x32) @ Wloc^T(32x128) ----
  __syncthreads();
  #pragma unroll
  for (int i = 0; i < 8; ++i) {         // As: 64x32
    int idx = tid + i * 256;
    int row = idx >> 5, f = idx & 31;
    int t = t0 + row;
    As[row * LSTRIDE + f] = (t < TT)
        ? locfeat[((size_t)b * TT + t) * NF + f] : 0.f;
  }
  #pragma unroll
  for (int i = 0; i < 16; ++i) {        // Bs[n][f] = Wloc[n][f]: 128x32
    int idx = tid + i * 256;
    int n = idx >> 5, f = idx & 31;
    Bs[n * LSTRIDE + f] = Wloc[n * NF + f];
  }
  __syncthreads();
  for (int kb = 0; kb < NF; kb += 4) {
    v2f bvec = *reinterpret_cast<const v2f*>(&Bs[(n0 + lm) * LSTRIDE + kb + kh]);
    #pragma unroll
    for (int tile = 0; tile < 4; ++tile) {
      v2f avec = *reinterpret_cast<const v2f*>(&As[(tile * 16 + lm) * LSTRIDE + kb + kh]);
      acc[tile] = __builtin_amdgcn_wmma_f32_16x16x4_f32(
          false, avec, false, bvec, (short)0, acc[tile], false, false);
    }
  }

  // ---- epilogue: tanh, weight by Wv, reduce over the 128 A-dim cols ----
  const int  n   = n0 + lm;             // this lane's output column
  const float qv  = qcomb[b * AD + n];
  const float wvn = Wv[n];
  #pragma unroll
  for (int tile = 0; tile < 4; ++tile) {
    #pragma unroll
    for (int r = 0; r < 8; ++r) {
      // C/D layout: VGPR r -> row tile*16 + r (lanes 0-15) or +8 (lanes 16-31)
      float e = tanhf(acc[tile][r] + qv) * wvn;
      e += __shfl_xor(e, 1, 32);
      e += __shfl_xor(e, 2, 32);
      e += __shfl_xor(e, 4, 32);
      e += __shfl_xor(e, 8, 32);
      if (lm == 0) {
        int row = tile * 16 + r + ((lane >> 4) << 3);
        atomicAdd(&ener[row], e);       // ds_add_f32; cross-wave col reduction
      }
    }
  }
  __syncthreads();
  if (tid < 64) {
    int t = t0 + tid;
    if (t < TT) energies[b * TPAD + t] = ener[tid];
  }
}

// ---------------- Stage 3: masked softmax over T per batch -----------------
__global__ __launch_bounds__(256) void softmax_kernel(
    const float* __restrict__ energies, const unsigned char* __restrict__ mask,
    float* __restrict__ weights)
{
  __shared__ float red[256];
  const int b = blockIdx.x;
  const int tid = threadIdx.x;
  float ev[4];
  float m = -1e30f;
  #pragma unroll
  for (int i = 0; i < 4; ++i) {
    int t = tid + i * 256;
    float e = -1e30f;
    if (t < TT) {
      e = energies[b * TPAD + t];
      if (mask[b * TT + t]) e = -1e30f;
    }
    ev[i] = e;
    m = fmaxf(m, e);
  }
  red[tid] = m; __syncthreads();
  for (int s = 128; s > 0; s >>= 1) {
    if (tid < s) red[tid] = fmaxf(red[tid], red[tid + s]);
    __syncthreads();
  }
  m = red[0]; __syncthreads();
  float xv[4]; float sum = 0.f;
  #pragma unroll
  for (int i = 0; i < 4; ++i) {
    int t = tid + i * 256;
    float x = (t < TT) ? expf(ev[i] - m) : 0.f;
    xv[i] = x; sum += x;
  }
  red[tid] = sum; __syncthreads();
  for (int s = 128; s > 0; s >>= 1) {
    if (tid < s) red[tid] += red[tid + s];
    __syncthreads();
  }
  float inv = 1.f / red[0];
  #pragma unroll
  for (int i = 0; i < 4; ++i) {
    int t = tid + i * 256;
    if (t < TT) weights[b * TT + t] = xv[i] * inv;
  }
}

// ---------------- Stage 4: context[b,m] = sum_t w[b,t]*memory[b,t,m] -------
__global__ __launch_bounds__(512) void context_kernel(
    const float* __restrict__ weights, const float* __restrict__ memory,
    float* __restrict__ context)
{
  __shared__ float wsh[TT];
  const int b = blockIdx.x;
  const int tid = threadIdx.x;
  for (int i = tid; i < TT; i += 512) wsh[i] = weights[b * TT + i];
  __syncthreads();
  const float* mrow = &memory[(size_t)b * TT * MD + tid];
  float acc = 0.f;
  for (int t = 0; t < TT; ++t) acc += wsh[t] * mrow[(size_t)t * MD];
  context[b * MD + tid] = acc;
}

extern "C" void kernel_launch(void* const* d_in, const int* in_sizes, int n_in,
                              void* d_out, int out_size, void* d_ws, size_t ws_size,
                              hipStream_t stream) {
  const float* query  = (const float*)d_in[0];
  const float* memory = (const float*)d_in[1];
  const float* awc    = (const float*)d_in[2];
  const unsigned char* mask = (const unsigned char*)d_in[3];
  const float* Wq     = (const float*)d_in[4];
  const float* bq     = (const float*)d_in[5];
  const float* Wm     = (const float*)d_in[6];
  const float* bm     = (const float*)d_in[7];
  const float* Wv     = (const float*)d_in[8];
  const float* bvp    = (const float*)d_in[9];
  const float* conv_w = (const float*)d_in[10];
  const float* conv_b = (const float*)d_in[11];
  const float* Wloc   = (const float*)d_in[12];

  float* out_ctx = (float*)d_out;            // [64, 512]
  float* out_w   = out_ctx + BB * MD;        // [64, 1000]

  float* locfeat  = (float*)d_ws;                          // 64*1000*32 f32
  float* qcomb    = locfeat + (size_t)BB * TT * NF;        // 64*128
  float* energies = qcomb + BB * AD;                       // 64*1024

  conv_kernel   <<<dim3((TT + 255) / 256, BB), 256, 0, stream>>>(awc, conv_w, locfeat);
  qproj_kernel  <<<dim3(BB), 128, 0, stream>>>(query, Wq, bq, bm, conv_b, Wloc, qcomb);
  energies_kernel<<<dim3((TT + 63) / 64, BB), 256, 0, stream>>>(
      memory, Wm, locfeat, Wloc, qcomb, Wv, bvp, energies);
  softmax_kernel<<<dim3(BB), 256, 0, stream>>>(energies, mask, out_w);
  context_kernel<<<dim3(BB), 512, 0, stream>>>(out_w, memory, out_ctx);
}